// SlicedEarthMoversDistance_4982162063681
// MI455X (gfx1250) — compile-verified
//
#include <hip/hip_runtime.h>
#include <math.h>

#define BATCH 8
#define CHN   3
#define HH    128
#define WW    128
#define HW    16384
#define NPROJ 100
#define NTILE 7        // ceil(100/16)
#define MTILES 1024    // 16384/16
#define EPSV  1e-6f

typedef __attribute__((ext_vector_type(2))) float v2f;
typedef __attribute__((ext_vector_type(8))) float v8f;

// Wave-relative LDS byte offset: low 32 bits of a generic pointer into LDS.
__device__ __forceinline__ unsigned lds_off(const void* p) {
    return (unsigned)(size_t)p;
}

// ---------------- init: zero accumulators ----------------
__global__ void k_init(float* out, float* sumX, float* sumY) {
    int t = threadIdx.x;
    if (t == 0) out[0] = 0.0f;
    if (t < BATCH) { sumX[t] = 0.0f; sumY[t] = 0.0f; }
}

// ---------------- EPS fixup: per-batch max/argmax of X channel 0 ----------------
__global__ void k_fix(const float* X, int* fixIdx) {
    __shared__ float sv[256];
    __shared__ int   sidx[256];
    int b = blockIdx.x, t = threadIdx.x;
    const float* x0 = X + (size_t)b * CHN * HW;
    float mv = -1.0f; int mi = 0;
    for (int k = t; k < HW; k += 256) {
        float v = x0[k];
        if (v > mv) { mv = v; mi = k; }
    }
    sv[t] = mv; sidx[t] = mi;
    __syncthreads();
    for (int s = 128; s > 0; s >>= 1) {
        if (t < s) {
            if (sv[t + s] > sv[t] || (sv[t + s] == sv[t] && sidx[t + s] < sidx[t])) {
                sv[t] = sv[t + s]; sidx[t] = sidx[t + s];
            }
        }
        __syncthreads();
    }
    if (t == 0) fixIdx[b] = (sv[0] < EPSV) ? sidx[0] : -1;
}

// ---------------- weights: wX = channel-mean(X) (with fixup), wY = Y[:,0]; batch sums ----------------
__global__ void k_weights(const float* X, const float* Y, const int* fixIdx,
                          float* wxr, float* wyr, float* sumX, float* sumY) {
    __shared__ float ru[256], rv[256];
    int g = blockIdx.x, t = threadIdx.x;
    int b = g >> 6;
    int k = ((g & 63) << 8) + t;
    const float* xb = X + (size_t)b * CHN * HW;
    float x0 = xb[k];
    if (k == fixIdx[b]) x0 = EPSV;
    float xm = (x0 + xb[HW + k] + xb[2 * HW + k]) * (1.0f / 3.0f);
    float yv = Y[(size_t)b * CHN * HW + k];
    wxr[b * HW + k] = xm;
    wyr[b * HW + k] = yv;
    ru[t] = xm; rv[t] = yv;
    __syncthreads();
    for (int s = 128; s > 0; s >>= 1) {
        if (t < s) { ru[t] += ru[t + s]; rv[t] += rv[t + s]; }
        __syncthreads();
    }
    if (t == 0) { atomicAdd(&sumX[b], ru[0]); atomicAdd(&sumY[b], rv[0]); }
}

// ---------------- projection GEMM via WMMA f32 16x16x4 ----------------
// pcT layout: [proj][pixel], pcT[n*HW + k] = i(k)*pn0[n] + j(k)*pn1[n]
__global__ void k_proj(const float* proj, float* pcT) {
    int wid  = (blockIdx.x * blockDim.x + threadIdx.x) >> 5;  // one wave = one 16x16 tile
    int lane = threadIdx.x & 31;
    int mT = wid / NTILE, nT = wid % NTILE;
    int Mb = mT << 4, Nb = nT << 4;
    int  l15 = lane & 15;
    bool lo  = lane < 16;

    // A fragment (16x4): lanes 0-15 hold K=0 (VGPR0) / K=1 (VGPR1); lanes 16-31 hold zero K-pad
    // B fragment (4x16): lanes 0-15 hold K=0 / K=1 rows for column N=Nb+lane; hi lanes zero pad
    v2f a = {0.0f, 0.0f};
    v2f bm = {0.0f, 0.0f};
    if (lo) {
        int k = Mb + l15;
        a[0] = (float)(k >> 7);    // i coordinate
        a[1] = (float)(k & 127);   // j coordinate
        int n = Nb + l15;
        if (n < NPROJ) {
            float p0 = proj[n], p1 = proj[NPROJ + n];
            float inv = 1.0f / sqrtf(p0 * p0 + p1 * p1);
            bm[0] = p0 * inv; bm[1] = p1 * inv;
        }
    }
    v8f c = {0.0f, 0.0f, 0.0f, 0.0f, 0.0f, 0.0f, 0.0f, 0.0f};
    v8f d;
#if __has_builtin(__builtin_amdgcn_wmma_f32_16x16x4_f32)
    d = __builtin_amdgcn_wmma_f32_16x16x4_f32(false, a, false, bm, (short)0, c,
                                              false, false);
#else
    // VALU fallback: compute this lane's D column fragment directly
    {
        int n = Nb + l15;
        float p0 = 0.0f, p1 = 0.0f;
        if (n < NPROJ) {
            float q0 = proj[n], q1 = proj[NPROJ + n];
            float inv = 1.0f / sqrtf(q0 * q0 + q1 * q1);
            p0 = q0 * inv; p1 = q1 * inv;
        }
        #pragma unroll
        for (int r = 0; r < 8; ++r) {
            int m = Mb + (lo ? r : r + 8);
            d[r] = (float)(m >> 7) * p0 + (float)(m & 127) * p1;
        }
    }
#endif
    // D layout: VGPR r -> (M=r, N=lane) for lanes 0-15; (M=r+8, N=lane-16) for lanes 16-31
    int n = Nb + l15;
    if (n < NPROJ) {
        float* dst = pcT + (size_t)n * HW + Mb + (lo ? 0 : 8);
        #pragma unroll
        for (int r = 0; r < 8; ++r) dst[r] = d[r];
    }
}

// ---------------- per-projection bitonic argsort of 16384 keys in LDS ----------------
// Keys staged global->LDS with async B128 copies (ASYNCcnt), results written back
// with async LDS->global stores (covered by S_ENDPGM implicit wait-idle).
__global__ void __launch_bounds__(1024) k_sort(float* pcT, int* sorter) {
    __shared__ float sk[HW];
    __shared__ int   si[HW];
    int p = blockIdx.x, t = threadIdx.x;
    float* col = pcT + (size_t)p * HW;

    {
        unsigned lk = lds_off(sk) + (unsigned)t * 16u;   // 16B per thread per chunk
        unsigned go = (unsigned)t * 16u;                 // global byte offset
        asm volatile(
            "global_load_async_to_lds_b128 %0, %1, %2\n\t"
            "global_load_async_to_lds_b128 %0, %1, %2 offset:16384\n\t"
            "global_load_async_to_lds_b128 %0, %1, %2 offset:32768\n\t"
            "global_load_async_to_lds_b128 %0, %1, %2 offset:49152"
            :: "v"(lk), "v"(go), "s"(col)
            : "memory");
    }
    for (int w = 0; w < 16; ++w) {
        int e = w * 1024 + t;
        si[e] = e;
    }
    asm volatile("s_wait_asynccnt 0x0" ::: "memory");
    __syncthreads();

    for (unsigned size = 2; size <= HW; size <<= 1) {
        for (unsigned stride = size >> 1; stride > 0; stride >>= 1) {
            __syncthreads();
            #pragma unroll
            for (int w = 0; w < 8; ++w) {
                unsigned q   = (unsigned)(w * 1024 + t);
                unsigned low = ((q & ~(stride - 1)) << 1) | (q & (stride - 1));
                unsigned hi  = low + stride;
                bool asc = ((low & size) == 0);
                float ka = sk[low], kb = sk[hi];
                if ((ka > kb) == asc) {
                    int ia = si[low];
                    sk[low] = kb; sk[hi] = ka;
                    si[low] = si[hi]; si[hi] = ia;
                }
            }
        }
    }
    __syncthreads();

    {
        int* so = sorter + (size_t)p * HW;
        unsigned lk = lds_off(sk) + (unsigned)t * 16u;
        unsigned li = lds_off(si) + (unsigned)t * 16u;
        unsigned go = (unsigned)t * 16u;
        asm volatile(
            "global_store_async_from_lds_b128 %2, %0, %3\n\t"
            "global_store_async_from_lds_b128 %2, %0, %3 offset:16384\n\t"
            "global_store_async_from_lds_b128 %2, %0, %3 offset:32768\n\t"
            "global_store_async_from_lds_b128 %2, %0, %3 offset:49152\n\t"
            "global_store_async_from_lds_b128 %2, %1, %4\n\t"
            "global_store_async_from_lds_b128 %2, %1, %4 offset:16384\n\t"
            "global_store_async_from_lds_b128 %2, %1, %4 offset:32768\n\t"
            "global_store_async_from_lds_b128 %2, %1, %4 offset:49152"
            :: "v"(lk), "v"(li), "v"(go), "s"(col), "s"(so)
            : "memory");
        // S_ENDPGM performs an implicit wait-idle (all counters, incl. ASYNCcnt)
    }
}

// ---------------- per-(batch,projection) W1 via dual cumsum + CDF integral ----------------
__global__ void __launch_bounds__(1024) k_swd(const float* pcT, const int* sorter,
                                              const float* wxr, const float* wyr,
                                              const float* sumX, const float* sumY,
                                              float* out) {
    __shared__ float swx[HW];
    __shared__ float swy[HW];
    __shared__ float scu[1024];
    __shared__ float scv[1024];
    int task = blockIdx.x;
    int b = task / NPROJ, p = task % NPROJ;
    int t = threadIdx.x;

    const float* wxb = wxr + (size_t)b * HW;
    const float* wyb = wyr + (size_t)b * HW;
    {
        unsigned lx = lds_off(swx) + (unsigned)t * 16u;
        unsigned ly = lds_off(swy) + (unsigned)t * 16u;
        unsigned go = (unsigned)t * 16u;
        asm volatile(
            "global_load_async_to_lds_b128 %0, %2, %3\n\t"
            "global_load_async_to_lds_b128 %0, %2, %3 offset:16384\n\t"
            "global_load_async_to_lds_b128 %0, %2, %3 offset:32768\n\t"
            "global_load_async_to_lds_b128 %0, %2, %3 offset:49152\n\t"
            "global_load_async_to_lds_b128 %1, %2, %4\n\t"
            "global_load_async_to_lds_b128 %1, %2, %4 offset:16384\n\t"
            "global_load_async_to_lds_b128 %1, %2, %4 offset:32768\n\t"
            "global_load_async_to_lds_b128 %1, %2, %4 offset:49152\n\t"
            "s_wait_asynccnt 0x0"
            :: "v"(lx), "v"(ly), "v"(go), "s"(wxb), "s"(wyb)
            : "memory");
    }
    __syncthreads();

    const int* so = sorter + (size_t)p * HW;
    int base = t * 16;
    float lu[16], lv[16];
    float ru = 0.0f, rv = 0.0f;
    #pragma unroll
    for (int i = 0; i < 16; ++i) {
        int s = so[base + i];
        ru += swx[s]; rv += swy[s];
        lu[i] = ru; lv[i] = rv;
    }
    scu[t] = ru; scv[t] = rv;
    __syncthreads();
    for (int off = 1; off < 1024; off <<= 1) {
        float au = 0.0f, av = 0.0f;
        if (t >= off) { au = scu[t - off]; av = scv[t - off]; }
        __syncthreads();
        scu[t] += au; scv[t] += av;
        __syncthreads();
    }
    float pu = (t > 0) ? scu[t - 1] : 0.0f;
    float pv = (t > 0) ? scv[t - 1] : 0.0f;
    float invX = 1.0f / sumX[b];
    float invY = 1.0f / sumY[b];

    const float* pcs = pcT + (size_t)p * HW;
    float acc = 0.0f;
    float x0 = pcs[base];
    #pragma unroll
    for (int i = 0; i < 16; ++i) {
        int k = base + i;
        if (k < HW - 1) {
            float x1 = pcs[k + 1];
            acc += fabsf((pu + lu[i]) * invX - (pv + lv[i]) * invY) * (x1 - x0);
            x0 = x1;
        }
    }
    __syncthreads();
    scu[t] = acc;
    __syncthreads();
    for (int s = 512; s > 0; s >>= 1) {
        if (t < s) scu[t] += scu[t + s];
        __syncthreads();
    }
    if (t == 0) atomicAdd(out, scu[0] * (1.0f / NPROJ));
}

extern "C" void kernel_launch(void* const* d_in, const int* in_sizes, int n_in,
                              void* d_out, int out_size, void* d_ws, size_t ws_size,
                              hipStream_t stream) {
    const float* X    = (const float*)d_in[0];
    const float* Y    = (const float*)d_in[1];
    const float* proj = (const float*)d_in[2];
    float* out = (float*)d_out;

    char* ws = (char*)d_ws;
    float* pcT    = (float*)ws;                                     // NPROJ*HW f32
    int*   sorter = (int*)(ws + (size_t)NPROJ * HW * 4);            // NPROJ*HW i32
    float* wxr    = (float*)(ws + (size_t)2 * NPROJ * HW * 4);      // BATCH*HW f32
    float* wyr    = wxr + (size_t)BATCH * HW;                       // BATCH*HW f32
    float* sumX   = wyr + (size_t)BATCH * HW;                       // BATCH f32
    float* sumY   = sumX + BATCH;                                   // BATCH f32
    int*   fixIdx = (int*)(sumY + BATCH);                           // BATCH i32

    k_init<<<1, 64, 0, stream>>>(out, sumX, sumY);
    k_fix<<<BATCH, 256, 0, stream>>>(X, fixIdx);
    k_weights<<<BATCH * 64, 256, 0, stream>>>(X, Y, fixIdx, wxr, wyr, sumX, sumY);
    k_proj<<<(MTILES * NTILE) / 8, 256, 0, stream>>>(proj, pcT);    // 8 waves/block, 1 tile/wave
    k_sort<<<NPROJ, 1024, 0, stream>>>(pcT, sorter);
    k_swd<<<BATCH * NPROJ, 1024, 0, stream>>>(pcT, sorter, wxr, wyr, sumX, sumY, out);
}